// SetAbstractionModule_56916906606961
// MI455X (gfx1250) — compile-verified
//
#include <hip/hip_runtime.h>
#include <stdint.h>

#define BB 8
#define NN 4096
#define SS 1024
#define KK 64
#define CIN 64
#define COUT 128
#define MM (BB*SS*KK)        /* 524288 edge rows */
#define RAD2 0.04f
#define EPSBN 1e-5f

typedef __attribute__((ext_vector_type(2))) float v2f;
typedef __attribute__((ext_vector_type(8))) float v8f;

// ---------------- order-preserving float<->uint encoding for atomicMax ----
__device__ __forceinline__ unsigned encodeF(float x){
  int b = __float_as_int(x);
  unsigned u = (unsigned)b;
  return (b >= 0) ? (u | 0x80000000u) : ~u;
}
__device__ __forceinline__ float decodeF(unsigned u){
  return (u & 0x80000000u) ? __int_as_float((int)(u & 0x7FFFFFFFu))
                           : __int_as_float((int)~u);
}

// ---------------- workspace zero-init (stats + max buffer) ----------------
__global__ void init_kernel(unsigned* __restrict__ p, int nwords){
  for (int i = blockIdx.x*256 + threadIdx.x; i < nwords; i += gridDim.x*256)
    p[i] = 0u;
}

// ---------------- FPS: one block per cloud, deterministic start at 0 ------
__global__ void fps_kernel(const float* __restrict__ pos, int* __restrict__ idxOut){
  const int b = blockIdx.x;
  const int tid = threadIdx.x;
  const int lane = tid & 31, wv = tid >> 5;
  const float* pb = pos + (size_t)b * NN * 3;
  const int PPT = NN / 256;                 // 16 points per thread
  float px[PPT], py[PPT], pz[PPT], md[PPT];
  __shared__ float sQ[3];
  __shared__ float sD[8];
  __shared__ int   sI[8];
  float q0x = pb[0], q0y = pb[1], q0z = pb[2];
#pragma unroll
  for (int i = 0; i < PPT; ++i){
    int p = tid + i*256;
    px[i] = pb[p*3+0]; py[i] = pb[p*3+1]; pz[i] = pb[p*3+2];
    float dx = px[i]-q0x, dy = py[i]-q0y, dz = pz[i]-q0z;
    md[i] = dx*dx + dy*dy + dz*dz;
  }
  if (tid == 0) idxOut[b*SS + 0] = 0;
  for (int s = 1; s < SS; ++s){
    float bd = -1.f; int bi = 0;
#pragma unroll
    for (int i = 0; i < PPT; ++i){
      int p = tid + i*256;
      if (md[i] > bd){ bd = md[i]; bi = p; }
    }
#pragma unroll
    for (int off = 16; off > 0; off >>= 1){
      float od = __shfl_xor(bd, off, 32);
      int   oi = __shfl_xor(bi, off, 32);
      if (od > bd || (od == bd && oi < bi)){ bd = od; bi = oi; }
    }
    if (lane == 0){ sD[wv] = bd; sI[wv] = bi; }
    __syncthreads();
    if (tid == 0){
      float best = sD[0]; int bidx = sI[0];
      for (int w = 1; w < 8; ++w)
        if (sD[w] > best || (sD[w] == best && sI[w] < bidx)){ best = sD[w]; bidx = sI[w]; }
      idxOut[b*SS + s] = bidx;
      sQ[0] = pb[bidx*3+0]; sQ[1] = pb[bidx*3+1]; sQ[2] = pb[bidx*3+2];
    }
    __syncthreads();
    float qx = sQ[0], qy = sQ[1], qz = sQ[2];
#pragma unroll
    for (int i = 0; i < PPT; ++i){
      float dx = px[i]-qx, dy = py[i]-qy, dz = pz[i]-qz;
      md[i] = fminf(md[i], dx*dx + dy*dy + dz*dz);
    }
  }
}

// ---------------- ball query (first-K by index) + gather ------------------
// one wave32 per centroid; feat row layout: [x_j(64) | rel(3) | 0] stride 68
__global__ void gather_kernel(const float* __restrict__ x, const float* __restrict__ pos,
                              const int* __restrict__ idxArr, float* __restrict__ feat,
                              unsigned char* __restrict__ maskArr, unsigned* __restrict__ cntPtr){
  __shared__ int sNbr[8*KK];
  const int lane = threadIdx.x & 31;
  const int wv = threadIdx.x >> 5;
  const int bs = blockIdx.x * 8 + wv;
  const int b  = bs / SS;
  volatile int* nbr = sNbr + wv*KK;
  const int ci = idxArr[bs];
  const float* pb = pos + (size_t)b*NN*3;
  const float qx = pb[ci*3+0], qy = pb[ci*3+1], qz = pb[ci*3+2];
  int found = 0;
  for (int base = 0; base < NN; base += 32){
    int p = base + lane;
    float dx = pb[p*3+0]-qx, dy = pb[p*3+1]-qy, dz = pb[p*3+2]-qz;
    bool pred = (dx*dx + dy*dy + dz*dz) <= RAD2;
    unsigned m = (unsigned)__ballot(pred);
    if (pred){
      int slot = found + __popc(m & ((1u << lane) - 1u));
      if (slot < KK) nbr[slot] = p;
    }
    found += __popc(m);
    if (found >= KK) break;
  }
  const int cnt = found < KK ? found : KK;
  if (lane == 0) atomicAdd(cntPtr, (unsigned)cnt);
  __threadfence_block();
  const float* xb = x + (size_t)b*NN*CIN;
  for (int slot = 0; slot < KK; ++slot){
    const int rowi = bs*KK + slot;
    float* fr = feat + (size_t)rowi*68;
    if (slot < cnt){
      int j = nbr[slot];
      const float* xr = xb + (size_t)j*CIN;
#pragma unroll
      for (int c = lane; c < CIN; c += 32) fr[c] = xr[c];
      if (lane < 3){
        float qc = (lane == 0) ? qx : ((lane == 1) ? qy : qz);
        fr[CIN + lane] = pb[j*3 + lane] - qc;
      }
      if (lane == 3) fr[67] = 0.f;
      if (lane == 0) maskArr[rowi] = 1;
    } else {
      for (int c = lane; c < 68; c += 32) fr[c] = 0.f;
      if (lane == 0) maskArr[rowi] = 0;
    }
  }
}

// ---------------- fused GEMM + (input BN/ReLU) + stats + (max) ------------
// h[M,NOUT] = act(A)[M,KD] @ W[KD(WREAL valid),NOUT] + bias
// 256 threads = 8 waves; wave -> one 16x16 WMMA output tile.
template<int KD, int WREAL, int NOUT, bool BN_IN, bool DO_MAX>
__global__ void gemm_kernel(const float* __restrict__ A,
                            const float* __restrict__ W,
                            const float* __restrict__ bias,
                            const float* __restrict__ bnA,
                            const float* __restrict__ bnC,
                            const unsigned char* __restrict__ mask,
                            float* __restrict__ out,
                            unsigned* __restrict__ maxEnc,
                            float* __restrict__ sumG,
                            float* __restrict__ sqG){
  constexpr int COLT = NOUT / 16;
  constexpr int ROWT = 8 / COLT;
  constexpr int ROWS = ROWT * 16;
  constexpr int WLD  = NOUT + 1;          // LDS pad vs bank conflicts
  constexpr int ALD  = 68;                // stride 68 -> conflict-free A reads
  __shared__ float sW[KD * WLD];
  __shared__ float sA[ROWS * ALD];
  __shared__ float sBias[NOUT];
  __shared__ float sBnA[BN_IN ? KD : 1];
  __shared__ float sBnC[BN_IN ? KD : 1];
  __shared__ unsigned char sMask[ROWS];
  __shared__ float sStat[2 * NOUT];
  const int tid  = threadIdx.x;
  const int lane = tid & 31;
  const int wv   = tid >> 5;
  const int rt   = wv / COLT;
  const int ct   = wv % COLT;
  const int col  = ct*16 + (lane & 15);
  const int khalf = (lane >> 4) << 1;     // lanes 0-15 -> K+0/1, lanes 16-31 -> K+2/3

  for (int i = tid; i < KD*NOUT; i += 256){
    int k = i / NOUT, n = i % NOUT;
    sW[k*WLD + n] = (k < WREAL) ? W[k*NOUT + n] : 0.f;
  }
  for (int i = tid; i < NOUT; i += 256) sBias[i] = bias[i];
  if (BN_IN)
    for (int i = tid; i < KD; i += 256){ sBnA[i] = bnA[i]; sBnC[i] = bnC[i]; }
  for (int i = tid; i < 2*NOUT; i += 256) sStat[i] = 0.f;

  float tSum = 0.f, tSq = 0.f;
  const int ntiles = MM / ROWS;
  for (int t = blockIdx.x; t < ntiles; t += gridDim.x){
    __syncthreads();
    for (int i = tid; i < ROWS*KD; i += 256){
      int r = i / KD, k = i % KD;
      float v = A[((size_t)t*ROWS + r)*KD + k];
      if (BN_IN) v = fmaxf(sBnA[k]*v + sBnC[k], 0.f);
      sA[r*ALD + k] = v;
    }
    if (tid < ROWS) sMask[tid] = mask[(size_t)t*ROWS + tid];
    __syncthreads();

    v8f acc = {};
#pragma unroll
    for (int k = 0; k < KD; k += 4){
      v2f av, bv;
      const float* ap = &sA[(rt*16 + (lane & 15))*ALD + k + khalf];
      av.x = ap[0];
      av.y = ap[1];
      const float* bp = &sW[(k + khalf)*WLD + col];
      bv.x = bp[0];
      bv.y = bp[WLD];
      acc = __builtin_amdgcn_wmma_f32_16x16x4_f32(false, av, false, bv,
                                                  (short)0, acc, false, false);
    }

    const float bcol = sBias[col];
    float mval = -__builtin_inff();
#pragma unroll
    for (int i = 0; i < 8; ++i){
      float h = acc[i] + bcol;
      int rlocal = rt*16 + i + ((lane >> 4) << 3);   // C/D layout: vgpr i -> M=i / M=i+8
      bool valid = sMask[rlocal] != 0;
      if (valid){
        tSum += h; tSq += h*h;
        if (DO_MAX) mval = fmaxf(mval, h);
      }
      if (!DO_MAX){
        out[((size_t)t*ROWS + rlocal)*NOUT + col] = h;
      }
    }
    if (DO_MAX){
      float o = __shfl_xor(mval, 16, 32);            // lanes l / l+16 share a column
      mval = fmaxf(mval, o);
      if (lane < 16 && mval > -__builtin_inff()){
        int bsIdx = (t*ROWS) / KK;                   // ROWS=16 divides K=64 -> one group
        atomicMax(&maxEnc[(size_t)bsIdx*NOUT + col], encodeF(mval));
      }
    }
  }

  __syncthreads();
  atomicAdd(&sStat[col], tSum);
  atomicAdd(&sStat[NOUT + col], tSq);
  __syncthreads();
  if (tid < NOUT){
    atomicAdd(&sumG[tid], sStat[tid]);
    atomicAdd(&sqG[tid],  sStat[NOUT + tid]);
  }
}

// ---------------- fold masked BN stats into per-channel affine ------------
__global__ void finalize_kernel(const float* __restrict__ sum, const float* __restrict__ sq,
                                const unsigned* __restrict__ cnt,
                                const float* __restrict__ g, const float* __restrict__ be,
                                float* __restrict__ bnA, float* __restrict__ bnC, int n){
  int c = threadIdx.x;
  if (c < n){
    float fc   = (float)(*cnt);
    float mean = sum[c] / fc;
    float var  = sq[c] / fc - mean*mean;
    float a    = g[c] * rsqrtf(var + EPSBN);
    bnA[c] = a;
    bnC[c] = be[c] - mean * a;
  }
}

// ---------------- out = relu(a3 * max_k(h3) + c3)  (a3 > 0 for g=1) -------
__global__ void finalout_kernel(const unsigned* __restrict__ maxEnc,
                                const float* __restrict__ bnA, const float* __restrict__ bnC,
                                float* __restrict__ outF){
  int i = blockIdx.x*256 + threadIdx.x;    // exactly B*S*COUT threads
  int c = i % COUT;
  float v = decodeF(maxEnc[i]);
  outF[i] = fmaxf(bnA[c]*v + bnC[c], 0.f);
}

// ---------------- pos_out gather + batch_out ------------------------------
__global__ void posbatch_kernel(const float* __restrict__ pos, const int* __restrict__ idxArr,
                                float* __restrict__ posOut, long long* __restrict__ batchOut){
  int bs = blockIdx.x*256 + threadIdx.x;
  if (bs < BB*SS){
    int b  = bs / SS;
    int ci = idxArr[bs];
    const float* p = pos + ((size_t)b*NN + ci)*3;
    posOut[bs*3+0] = p[0]; posOut[bs*3+1] = p[1]; posOut[bs*3+2] = p[2];
    batchOut[bs] = (long long)b;
  }
}

// ---------------- workspace layout ----------------------------------------
static constexpr size_t OFF_MAXENC = 8192;                          // stats region first (8KB)
static constexpr size_t OFF_IDX    = OFF_MAXENC + (size_t)BB*SS*COUT*4;   // +4MB
static constexpr size_t OFF_MASK   = OFF_IDX + (size_t)BB*SS*4;
static constexpr size_t OFF_FEAT   = OFF_MASK + (size_t)MM;         // byte mask
static constexpr size_t OFF_H1     = OFF_FEAT + (size_t)MM*68*4;    // feat / h2 reuse

extern "C" void kernel_launch(void* const* d_in, const int* in_sizes, int n_in,
                              void* d_out, int out_size, void* d_ws, size_t ws_size,
                              hipStream_t stream){
  const float* x   = (const float*)d_in[0];
  const float* pos = (const float*)d_in[1];
  const float* W1  = (const float*)d_in[3];
  const float* b1  = (const float*)d_in[4];
  const float* g1  = (const float*)d_in[5];
  const float* be1 = (const float*)d_in[6];
  const float* W2  = (const float*)d_in[7];
  const float* b2  = (const float*)d_in[8];
  const float* g2  = (const float*)d_in[9];
  const float* be2 = (const float*)d_in[10];
  const float* W3  = (const float*)d_in[11];
  const float* b3  = (const float*)d_in[12];
  const float* g3  = (const float*)d_in[13];
  const float* be3 = (const float*)d_in[14];

  char* ws = (char*)d_ws;
  float*    stats   = (float*)ws;                 // [0..511] sums, [512..] bn, word 1024 = cnt
  unsigned* cnt     = (unsigned*)(stats + 1024);
  unsigned* maxEnc  = (unsigned*)(ws + OFF_MAXENC);
  int*      idxArr  = (int*)(ws + OFF_IDX);
  unsigned char* maskArr = (unsigned char*)(ws + OFF_MASK);
  float*    feat    = (float*)(ws + OFF_FEAT);
  float*    h1      = (float*)(ws + OFF_H1);
  float*    h2      = feat;                       // feat dead after GEMM1 -> reuse

  float*     outF     = (float*)d_out;
  float*     posOut   = outF + (size_t)BB*SS*COUT;
  long long* batchOut = (long long*)(posOut + (size_t)BB*SS*3);

  // zero stats region + max buffer each call (deterministic across replays)
  init_kernel<<<2048, 256, 0, stream>>>((unsigned*)ws, (int)((OFF_IDX) / 4));

  fps_kernel<<<BB, 256, 0, stream>>>(pos, idxArr);
  gather_kernel<<<BB*SS/8, 256, 0, stream>>>(x, pos, idxArr, feat, maskArr, cnt);

  // layer 1: feat[M,68] @ W1[67,64] -> h1 ; stats1
  gemm_kernel<68,67,64,false,false><<<4096, 256, 0, stream>>>(
      feat, W1, b1, nullptr, nullptr, maskArr, h1, nullptr, stats + 0, stats + 64);
  finalize_kernel<<<1, 64, 0, stream>>>(stats + 0, stats + 64, cnt, g1, be1,
                                        stats + 512, stats + 576, 64);

  // layer 2: relu(bn1(h1)) @ W2 -> h2 ; stats2
  gemm_kernel<64,64,64,true,false><<<4096, 256, 0, stream>>>(
      h1, W2, b2, stats + 512, stats + 576, maskArr, h2, nullptr, stats + 128, stats + 192);
  finalize_kernel<<<1, 64, 0, stream>>>(stats + 128, stats + 192, cnt, g2, be2,
                                        stats + 640, stats + 704, 64);

  // layer 3: relu(bn2(h2)) @ W3 -> masked max_k (encoded atomicMax) ; stats3
  gemm_kernel<64,64,128,true,true><<<4096, 256, 0, stream>>>(
      h2, W3, b3, stats + 640, stats + 704, maskArr, nullptr, maxEnc, stats + 256, stats + 384);
  finalize_kernel<<<1, 128, 0, stream>>>(stats + 256, stats + 384, cnt, g3, be3,
                                         stats + 768, stats + 896, 128);

  finalout_kernel<<<(BB*SS*COUT)/256, 256, 0, stream>>>(maxEnc, stats + 768, stats + 896, outF);
  posbatch_kernel<<<(BB*SS + 255)/256, 256, 0, stream>>>(pos, idxArr, posOut, batchOut);
}